// FHGN_59863254172283
// MI455X (gfx1250) — compile-verified
//
#include <hip/hip_runtime.h>
#include <stdint.h>

// FHGN spiking recurrence:  u_t = u_{t-1}*(1-s_{t-1})*0.5 + x_t ; s_t = H(u_t - 1)
// Pure streaming: 268 MB total @ 23.3 TB/s => ~11.5 us roofline floor.
// CDNA5 path: 4-deep GLOBAL_LOAD_ASYNC_TO_LDS_B128 ring (prefetch distance 3,
// ASYNCcnt-tracked, s_wait_asynccnt immediates constant-folded by full unroll),
// recurrence state held in VGPRs, spikes written with TH_STORE_NT b128 stores.

#define T_STEPS 64
#define VEC     4                 // floats per thread per timestep (one b128)
#define BLOCK   256               // 8 wave32s
#define TILE_F  (BLOCK * VEC)     // 1024 floats = 4 KB per tile
#define NBUF    4                 // LDS ring depth (16 KB)
#define DIST    3                 // prefetch distance

typedef float v4f __attribute__((ext_vector_type(4)));

__device__ __forceinline__ void async_tile_load(uint32_t lds_off, uint64_t gaddr) {
    asm volatile("global_load_async_to_lds_b128 %0, %1, off"
                 :: "v"(lds_off), "v"(gaddr) : "memory");
}

__global__ __launch_bounds__(BLOCK)
void fhgn_async_kernel(const float* __restrict__ x,
                       float* __restrict__ out,
                       int BN) {
    __shared__ float tile[NBUF * TILE_F];   // 16 KB ring buffer

    const int tid  = threadIdx.x;
    const int base = blockIdx.x * TILE_F + tid * VEC;

    if (base + VEC <= BN) {
        // Wave-relative LDS byte offsets of this thread's 16B slot per ring slot.
        uint32_t lds[NBUF];
        #pragma unroll
        for (int b = 0; b < NBUF; ++b)
            lds[b] = (uint32_t)(uintptr_t)(&tile[b * TILE_F + tid * VEC]);

        const float* gx = x + base;

        // Prime the ring: tiles 0..DIST-1 in flight.
        #pragma unroll
        for (int p = 0; p < DIST; ++p)
            async_tile_load(lds[p], (uint64_t)(uintptr_t)(gx + (size_t)p * (size_t)BN));

        float u0 = 0.f, u1 = 0.f, u2 = 0.f, u3 = 0.f;
        float s0 = 0.f, s1 = 0.f, s2 = 0.f, s3 = 0.f;

        #pragma unroll
        for (int t = 0; t < T_STEPS; ++t) {
            // Keep DIST tiles in flight.
            if (t + DIST < T_STEPS)
                async_tile_load(lds[(t + DIST) & (NBUF - 1)],
                                (uint64_t)(uintptr_t)(gx + (size_t)(t + DIST) * (size_t)BN));

            // Wait until only the tiles beyond t remain outstanding => tile t
            // complete (async loads complete in order per wave). t is a
            // compile-time constant (full unroll), so this folds to one wait.
            const int rem = T_STEPS - 1 - t;
            if (rem >= 3)      asm volatile("s_wait_asynccnt 0x3" ::: "memory");
            else if (rem == 2) asm volatile("s_wait_asynccnt 0x2" ::: "memory");
            else if (rem == 1) asm volatile("s_wait_asynccnt 0x1" ::: "memory");
            else               asm volatile("s_wait_asynccnt 0x0" ::: "memory");

            const float4 xv = *reinterpret_cast<const float4*>(
                &tile[(t & (NBUF - 1)) * TILE_F + tid * VEC]);

            // u = u*(1-s)*0.5 + x ; with s in {0,1}:  u = s ? x : fma(u,0.5,x)
            u0 = (s0 != 0.f) ? xv.x : fmaf(u0, 0.5f, xv.x);
            u1 = (s1 != 0.f) ? xv.y : fmaf(u1, 0.5f, xv.y);
            u2 = (s2 != 0.f) ? xv.z : fmaf(u2, 0.5f, xv.z);
            u3 = (s3 != 0.f) ? xv.w : fmaf(u3, 0.5f, xv.w);
            s0 = (u0 >= 1.f) ? 1.f : 0.f;
            s1 = (u1 >= 1.f) ? 1.f : 0.f;
            s2 = (u2 >= 1.f) ? 1.f : 0.f;
            s3 = (u3 >= 1.f) ? 1.f : 0.f;

            // Streaming output: bypass cache retention (TH_STORE_NT).
            v4f sv = (v4f){s0, s1, s2, s3};
            __builtin_nontemporal_store(
                sv, reinterpret_cast<v4f*>(out + (size_t)t * (size_t)BN + base));
        }
    } else if (base < BN) {
        // Scalar tail (never hit for T=64,B=128,N=4096; kept for safety).
        for (int e = base; e < BN && e < base + VEC; ++e) {
            float u = 0.f, s = 0.f;
            for (int t = 0; t < T_STEPS; ++t) {
                const float xt = x[(size_t)t * (size_t)BN + e];
                u = (s != 0.f) ? xt : fmaf(u, 0.5f, xt);
                s = (u >= 1.f) ? 1.f : 0.f;
                out[(size_t)t * (size_t)BN + e] = s;
            }
        }
    }
}

extern "C" void kernel_launch(void* const* d_in, const int* in_sizes, int n_in,
                              void* d_out, int out_size, void* d_ws, size_t ws_size,
                              hipStream_t stream) {
    const float* x   = (const float*)d_in[0];
    float*       out = (float*)d_out;

    const int n  = in_sizes[0];        // T*B*N
    const int BN = n / T_STEPS;        // elements per timestep (B*N)

    const int grid = (BN + TILE_F - 1) / TILE_F;   // 512 blocks for reference shape
    fhgn_async_kernel<<<grid, BLOCK, 0, stream>>>(x, out, BN);
}